// VOC2010AveragePrecision_20444044329047
// MI455X (gfx1250) — compile-verified
//
#include <hip/hip_runtime.h>
#include <hip/hip_bf16.h>

typedef float v4f __attribute__((ext_vector_type(4)));
typedef float v2f __attribute__((ext_vector_type(2)));
typedef float v8f __attribute__((ext_vector_type(8)));

// One wave32 per (image, class) row of 1024 confidences.
// Lane L owns elements [L*32, L*32+32).
__global__ __launch_bounds__(256) void
voc2010_ap_kernel(const float* __restrict__ prec,
                  const float* __restrict__ rec,
                  float* __restrict__ out,
                  int nrows)
{
    const int lane = threadIdx.x & 31;
    const int wave = threadIdx.x >> 5;
    const int row  = blockIdx.x * 8 + wave;
    if (row >= nrows) return;   // wave-uniform; EXEC stays all-ones

    const v4f* __restrict__ P4 =
        (const v4f*)(prec + (size_t)row * 1024) + lane * 8;
    const v4f* __restrict__ R4 =
        (const v4f*)(rec  + (size_t)row * 1024) + lane * 8;

    // ---- Stream this lane's 32-element chunk into registers (NT hint:
    // 336MB read-once stream > 192MB L2, don't pollute it). ----
    float p[32], r[32];
#pragma unroll
    for (int k = 0; k < 8; ++k) {
        v4f t = __builtin_nontemporal_load(P4 + k);
        v4f u = __builtin_nontemporal_load(R4 + k);
        p[4*k+0] = t.x; p[4*k+1] = t.y; p[4*k+2] = t.z; p[4*k+3] = t.w;
        r[4*k+0] = u.x; r[4*k+1] = u.y; r[4*k+2] = u.z; r[4*k+3] = u.w;
    }

    // ---- Per-lane chunk max (feeds the cross-lane suffix-max scan). ----
    float cmax = p[0];
#pragma unroll
    for (int i = 1; i < 32; ++i) cmax = fmaxf(cmax, p[i]);

    // ---- Exclusive suffix-max over lanes (max over lanes > L). ----
    float inc = cmax;
#pragma unroll
    for (int off = 1; off < 32; off <<= 1) {
        float v = __shfl_down(inc, off, 32);
        inc = (lane + off < 32) ? fmaxf(inc, v) : inc;
    }
    float sfx = __shfl_down(inc, 1, 32);
    if (lane == 31) sfx = -__builtin_inff();

    // ---- r[-1] for this chunk: last element of the left neighbor. ----
    float rprev = __shfl_up(r[31], 1, 32);
    if (lane == 0) rprev = 0.0f;

    // ---- Fused right-to-left pass: running suffix max + FMA integrate. ----
    float pm  = sfx;
    float acc = 0.0f;
#pragma unroll
    for (int i = 31; i >= 1; --i) {
        pm  = fmaxf(pm, p[i]);
        acc = fmaf(pm, r[i] - r[i-1], acc);
    }
    pm  = fmaxf(pm, p[0]);
    acc = fmaf(pm, r[0] - rprev, acc);

    // ---- Wave-sum of the 32 per-lane partials via the matrix pipe.
    // A(16x4): lane's partial in its first A slot, 0 in the second; B = 1.0
    // everywhere, so D[m][n] = sum_k A[m][k]. Summing D over M (8 regs per
    // lane + the lane^16 partner) = sum of all 32 partials, independent of
    // the exact K<->lane mapping. EXEC is all-ones here as WMMA requires. ----
    v2f a; a.x = acc;  a.y = 0.0f;
    v2f b; b.x = 1.0f; b.y = 1.0f;
    v8f c = {};
    v8f d = __builtin_amdgcn_wmma_f32_16x16x4_f32(
        /*neg_a=*/false, a, /*neg_b=*/false, b,
        /*c_mod=*/(short)0, c, /*reuse_a=*/false, /*reuse_b=*/false);

    float s = ((d[0] + d[1]) + (d[2] + d[3])) + ((d[4] + d[5]) + (d[6] + d[7]));
    s += __shfl_xor(s, 16, 32);

    if (lane == 0) out[row] = s;
}

extern "C" void kernel_launch(void* const* d_in, const int* in_sizes, int n_in,
                              void* d_out, int out_size, void* d_ws, size_t ws_size,
                              hipStream_t stream)
{
    const float* prec = (const float*)d_in[0];   // (512, 80, 1024) f32
    const float* rec  = (const float*)d_in[1];   // (512, 80, 1024) f32
    float*       out  = (float*)d_out;           // (512, 80) f32

    const int nrows = out_size;                  // 512*80 = 40960 rows
    const int waves_per_block = 8;               // 256 threads = 8 wave32
    const int blocks = (nrows + waves_per_block - 1) / waves_per_block;  // 5120

    voc2010_ap_kernel<<<blocks, waves_per_block * 32, 0, stream>>>(
        prec, rec, out, nrows);
}